// InvMPNN_1803886265808
// MI455X (gfx1250) — compile-verified
//
#include <hip/hip_runtime.h>
#include <cstdint>
#include <cstddef>

typedef float v2f __attribute__((ext_vector_type(2)));
typedef float v8f __attribute__((ext_vector_type(8)));

#define N_NODES 50000
#define DIM 128
#define E_EDGES 800000
#define S_STEPS 4
#define LN_EPS 1e-5f

// ---------------------------------------------------------------- utilities

__global__ void zero_f32(float* __restrict__ p, int n) {
  int i = blockIdx.x * blockDim.x + threadIdx.x;
  if (i < n) p[i] = 0.0f;
}

// dist[e] = ||pos[row]-pos[col]||, cnt[col] += 1
__global__ void edge_geom(const float* __restrict__ pos,
                          const long long* __restrict__ ei,
                          float* __restrict__ dist,
                          float* __restrict__ cnt) {
  int e = blockIdx.x * blockDim.x + threadIdx.x;
  if (e >= E_EDGES) return;
  long long r = ei[e];
  long long c = ei[E_EDGES + e];
  float dx = pos[r * 3 + 0] - pos[c * 3 + 0];
  float dy = pos[r * 3 + 1] - pos[c * 3 + 1];
  float dz = pos[r * 3 + 2] - pos[c * 3 + 2];
  dist[e] = sqrtf(dx * dx + dy * dy + dz * dz);
  unsafeAtomicAdd(&cnt[c], 1.0f);
}

// Swizzled LDS weight stage: for each (kc, n) store K-rows in order
// [4kc+0, 4kc+2, 4kc+1, 4kc+3] so a lane's B-fragment (rows 4kc+half,
// 4kc+2+half) is one aligned 8-byte ds_load_b64 into an adjacent VGPR pair.
__device__ __forceinline__ void stage_wblock(float* wlds, const float* gsrc,
                                             int tid) {
#pragma unroll
  for (int i = 0; i < 8; ++i) {
    int idx = tid + i * 256;          // 0..2047 : (kc, n) pairs
    int kc  = idx >> 7;               // 0..15
    int n   = idx & 127;
    const float* g = gsrc + (size_t)(kc * 4) * 128 + n;  // coalesced over n
    float4 w;
    w.x = g[0 * 128];
    w.y = g[2 * 128];
    w.z = g[1 * 128];
    w.w = g[3 * 128];
    ((float4*)wlds)[idx] = w;         // one ds_store_b128
  }
}

__device__ __forceinline__ v2f load_bfrag(const float* wlds, int kc, int nt,
                                          int nl, int half) {
  return *(const v2f*)(wlds + (((kc * 128 + nt * 16 + nl) << 2) + half * 2));
}

// Template-unrolled chain of 8 WMMAs sharing one A fragment; NT is a
// compile-time constant so the reuse-A hint (required ICE) is legal on
// instructions 2..8 of the chain.
template <int NT>
__device__ __forceinline__ void wmma_chain(v8f (&acc)[8], v2f a,
                                           const float* wlds, int kc,
                                           int nl, int half) {
  v2f b = load_bfrag(wlds, kc, NT, nl, half);
  acc[NT] = __builtin_amdgcn_wmma_f32_16x16x4_f32(
      false, a, false, b, (short)0, acc[NT], NT != 0, false);
  if constexpr (NT < 7) wmma_chain<NT + 1>(acc, a, wlds, kc, nl, half);
}

// ------------------------------------------------- fused edge-message kernel
// One wave = 16 edges x 128 features.  GEMM K = 256 via WMMA f32 16x16x4,
// distance column (K=256) as rank-1 epilogue.  GELU + LayerNorm fused,
// scatter with global f32 atomics into agg.
__launch_bounds__(256)
__global__ void edge_msg(const float* __restrict__ h,
                         const long long* __restrict__ ei,
                         const float* __restrict__ dist,
                         const float* __restrict__ msg_w,    // [S,257,128]
                         const float* __restrict__ msg_b,    // [S,128]
                         const float* __restrict__ msg_g,
                         const float* __restrict__ msg_beta,
                         float* __restrict__ agg,            // [N,128]
                         int step) {
  __shared__ __align__(16) float wlds[64 * 128];  // 32 KB K-block (swizzled)

  const int tid  = threadIdx.x;
  const int wave = tid >> 5;
  const int lane = tid & 31;
  const int nl   = lane & 15;   // N column within tile / A-row M
  const int half = lane >> 4;   // which 16-lane half

  const int tile   = blockIdx.x * 8 + wave;  // 16-edge tile
  const int e_base = tile * 16;              // E is a multiple of 16

  // A-operand gather bases for this lane's edge (M = nl)
  const long long er = ei[e_base + nl];
  const long long ec = ei[E_EDGES + e_base + nl];
  const float* __restrict__ arow = h + (size_t)er * DIM;
  const float* __restrict__ acol = h + (size_t)ec * DIM;

  v8f acc[8];
  const v8f vzero = {0.f, 0.f, 0.f, 0.f, 0.f, 0.f, 0.f, 0.f};
#pragma unroll
  for (int t = 0; t < 8; ++t) acc[t] = vzero;

  const float* wsrc = msg_w + (size_t)step * 257 * 128;

  for (int kb = 0; kb < 4; ++kb) {  // 4 K-blocks of 64 (K = 0..255)
    __syncthreads();
    stage_wblock(wlds, wsrc + (size_t)kb * 64 * 128, tid);
    __syncthreads();

    const float* abase = (kb < 2) ? (arow + (kb & 1) * 64)
                                  : (acol + (kb & 1) * 64);
    for (int kc = 0; kc < 16; ++kc) {  // 16 chunks of K=4
      // A frag: lanes 0-15 hold K = 4kc,4kc+1 ; lanes 16-31 hold 4kc+2,4kc+3
      v2f a = *(const v2f*)(abase + kc * 4 + half * 2);
      wmma_chain<0>(acc, a, wlds, kc, nl, half);
    }
  }

  // ------------------------------------------------------------- epilogue
  float bn[8], gn[8], btn[8], wn[8];
  const float* bptr = msg_b    + (size_t)step * DIM;
  const float* gptr = msg_g    + (size_t)step * DIM;
  const float* tptr = msg_beta + (size_t)step * DIM;
  const float* w256 = wsrc + (size_t)256 * 128;  // distance-column weights
#pragma unroll
  for (int nt = 0; nt < 8; ++nt) {
    int n = nt * 16 + nl;
    bn[nt]  = bptr[n];
    gn[nt]  = gptr[n];
    btn[nt] = tptr[n];
    wn[nt]  = w256[n];
  }

  float dj[8];
  long long cj[8];
#pragma unroll
  for (int j = 0; j < 8; ++j) {
    int e = e_base + half * 8 + j;  // C layout: VGPR j -> M=j (half 0), j+8 (half 1)
    dj[j] = dist[e];
    cj[j] = ei[E_EDGES + e];
  }

#pragma unroll
  for (int j = 0; j < 8; ++j) {
    float v[8];
    float s1 = 0.f, s2 = 0.f;
#pragma unroll
    for (int nt = 0; nt < 8; ++nt) {
      float x = acc[nt][j] + bn[nt] + dj[j] * wn[nt];
      // exact GELU
      x = 0.5f * x * (1.0f + erff(x * 0.70710678118654752f));
      v[nt] = x;
      s1 += x;
      s2 += x * x;
    }
    // LayerNorm over 128 features: 8 per lane x 16 lanes of this half
#pragma unroll
    for (int off = 1; off < 16; off <<= 1) {
      s1 += __shfl_xor(s1, off, 32);
      s2 += __shfl_xor(s2, off, 32);
    }
    float mean = s1 * (1.0f / 128.0f);
    float var  = s2 * (1.0f / 128.0f) - mean * mean;
    float rstd = rsqrtf(var + LN_EPS);
    float* aggrow = agg + (size_t)cj[j] * DIM;
#pragma unroll
    for (int nt = 0; nt < 8; ++nt) {
      float y = (v[nt] - mean) * rstd * gn[nt] + btn[nt];
      unsafeAtomicAdd(&aggrow[nt * 16 + nl], y);
    }
  }
}

// ------------------------------------------------- fused node-update kernel
// One wave = 16 nodes x 128 features.  K = 256: first half from h, second
// half from agg * (1/max(count,1)).  LayerNorm + residual fused.
__launch_bounds__(256)
__global__ void node_upd(float* __restrict__ hio,            // [N,128] in/out
                         const float* __restrict__ agg,      // [N,128]
                         const float* __restrict__ cnt,      // [N]
                         const float* __restrict__ upd_w,    // [S,256,128]
                         const float* __restrict__ upd_b,
                         const float* __restrict__ upd_g,
                         const float* __restrict__ upd_beta,
                         int step) {
  __shared__ __align__(16) float wlds[64 * 128];

  const int tid  = threadIdx.x;
  const int wave = tid >> 5;
  const int lane = tid & 31;
  const int nl   = lane & 15;
  const int half = lane >> 4;

  const int tile   = blockIdx.x * 8 + wave;
  const bool active = tile < (N_NODES / 16);  // N is a multiple of 16
  const int n_base = active ? tile * 16 : 0;

  const float* __restrict__ arow = hio + (size_t)(n_base + nl) * DIM;
  const float* __restrict__ acol = agg + (size_t)(n_base + nl) * DIM;
  const float rden = 1.0f / fmaxf(cnt[n_base + nl], 1.0f);

  v8f acc[8];
  const v8f vzero = {0.f, 0.f, 0.f, 0.f, 0.f, 0.f, 0.f, 0.f};
#pragma unroll
  for (int t = 0; t < 8; ++t) acc[t] = vzero;

  const float* wsrc = upd_w + (size_t)step * 256 * 128;

  for (int kb = 0; kb < 4; ++kb) {
    __syncthreads();
    stage_wblock(wlds, wsrc + (size_t)kb * 64 * 128, tid);
    __syncthreads();
    if (!active) continue;  // wave-uniform; next barrier still reached by all

    const float* abase = (kb < 2) ? (arow + kb * 64) : (acol + (kb - 2) * 64);
    const float scale  = (kb < 2) ? 1.0f : rden;
    for (int kc = 0; kc < 16; ++kc) {
      v2f a = *(const v2f*)(abase + kc * 4 + half * 2);
      a.x *= scale;
      a.y *= scale;
      wmma_chain<0>(acc, a, wlds, kc, nl, half);
    }
  }

  if (!active) return;

  float bn[8], gn[8], btn[8];
  const float* bptr = upd_b    + (size_t)step * DIM;
  const float* gptr = upd_g    + (size_t)step * DIM;
  const float* tptr = upd_beta + (size_t)step * DIM;
#pragma unroll
  for (int nt = 0; nt < 8; ++nt) {
    int n = nt * 16 + nl;
    bn[nt]  = bptr[n];
    gn[nt]  = gptr[n];
    btn[nt] = tptr[n];
  }

#pragma unroll
  for (int j = 0; j < 8; ++j) {
    const int node = n_base + half * 8 + j;
    float v[8];
    float s1 = 0.f, s2 = 0.f;
#pragma unroll
    for (int nt = 0; nt < 8; ++nt) {
      float x = acc[nt][j] + bn[nt];
      v[nt] = x;
      s1 += x;
      s2 += x * x;
    }
#pragma unroll
    for (int off = 1; off < 16; off <<= 1) {
      s1 += __shfl_xor(s1, off, 32);
      s2 += __shfl_xor(s2, off, 32);
    }
    float mean = s1 * (1.0f / 128.0f);
    float var  = s2 * (1.0f / 128.0f) - mean * mean;
    float rstd = rsqrtf(var + LN_EPS);
    float* hrow = hio + (size_t)node * DIM;
#pragma unroll
    for (int nt = 0; nt < 8; ++nt) {
      float u = (v[nt] - mean) * rstd * gn[nt] + btn[nt];
      hrow[nt * 16 + nl] += u;  // residual
    }
  }
}

// ---------------------------------------------------------------- launcher

extern "C" void kernel_launch(void* const* d_in, const int* in_sizes, int n_in,
                              void* d_out, int out_size, void* d_ws, size_t ws_size,
                              hipStream_t stream) {
  const float*     h0       = (const float*)d_in[0];
  const float*     pos      = (const float*)d_in[1];
  const long long* ei       = (const long long*)d_in[2];  // int64 edge_index
  const float*     msg_w    = (const float*)d_in[3];
  const float*     msg_b    = (const float*)d_in[4];
  const float*     msg_g    = (const float*)d_in[5];
  const float*     msg_beta = (const float*)d_in[6];
  const float*     upd_w    = (const float*)d_in[7];
  const float*     upd_b    = (const float*)d_in[8];
  const float*     upd_g    = (const float*)d_in[9];
  const float*     upd_beta = (const float*)d_in[10];

  float* h = (float*)d_out;

  // workspace layout
  char*  ws   = (char*)d_ws;
  float* agg  = (float*)ws;                                      // N*D floats
  float* cnt  = (float*)(ws + (size_t)N_NODES * DIM * 4);        // N floats
  float* dist = (float*)(ws + (size_t)N_NODES * DIM * 4 + (size_t)N_NODES * 4);  // E floats

  // h = scalar_features (work in-place on d_out)
  (void)hipMemcpyAsync(h, h0, (size_t)N_NODES * DIM * sizeof(float),
                       hipMemcpyDeviceToDevice, stream);

  zero_f32<<<(N_NODES + 255) / 256, 256, 0, stream>>>(cnt, N_NODES);
  edge_geom<<<(E_EDGES + 255) / 256, 256, 0, stream>>>(pos, ei, dist, cnt);

  for (int s = 0; s < S_STEPS; ++s) {
    zero_f32<<<(N_NODES * DIM + 255) / 256, 256, 0, stream>>>(agg, N_NODES * DIM);
    edge_msg<<<E_EDGES / 128, 256, 0, stream>>>(h, ei, dist, msg_w, msg_b,
                                                msg_g, msg_beta, agg, s);
    node_upd<<<(N_NODES / 16 + 7) / 8, 256, 0, stream>>>(h, agg, cnt, upd_w,
                                                         upd_b, upd_g,
                                                         upd_beta, s);
  }
}